// HierarchicalSparseAttentionTriton_42374147343070
// MI455X (gfx1250) — compile-verified
//
#include <hip/hip_runtime.h>
#include <hip/hip_bf16.h>
#include <math.h>

// B=2, N=2048, H=16, D=64. Output [B,N,H,D] fp32.
#define B_ 2
#define N_ 2048
#define H_ 16
#define D_ 64
#define NINT_ (N_ - 1) // internal tree nodes per batch (stored in d_ws)

typedef __attribute__((ext_vector_type(2))) float v2f;
typedef __attribute__((ext_vector_type(8))) float v8f;

// global (address-space 1) load helpers: force global_load_* instead of
// flat_load_* when the address was round-tripped through LDS as an integer.
typedef const float __attribute__((address_space(1)))* gfp;
typedef const v2f   __attribute__((address_space(1)))* gv2p;

__device__ __forceinline__ v2f  load_g2(unsigned long long a) { return *(gv2p)a; }
__device__ __forceinline__ float load_g1(unsigned long long a) { return *(gfp)a; }

// ---------------------------------------------------------------------------
// Phase 1: tree build, one level per launch.
// Parent = ws*Kp0 + w0*K0 + w1*K1 with Kp0 = 0.5(K0+K1); same weights for V.
// Folds to parent = c0*child0 + c1*child1 with c0 = 0.5*ws+w0, c1 = 0.5*ws+w1.
// ---------------------------------------------------------------------------
__global__ __launch_bounds__(256) void tree_level_kernel(
    const float* __restrict__ Kc, const float* __restrict__ Vc,
    float* __restrict__ Kp, float* __restrict__ Vp,
    size_t cStride, size_t pStride, int cOff, int pOff, int P)
{
  int tid = blockIdx.x * blockDim.x + threadIdx.x;
  int total = B_ * P * H_;
  if (tid >= total) return;
  int h = tid % H_;
  int i = (tid / H_) % P;
  int b = tid / (H_ * P);

  const float* k0 = Kc + (size_t)b * cStride + ((size_t)(cOff + 2 * i) * H_ + h) * D_;
  const float* k1 = k0 + (size_t)H_ * D_;
  const float* v0 = Vc + (size_t)b * cStride + ((size_t)(cOff + 2 * i) * H_ + h) * D_;
  const float* v1 = v0 + (size_t)H_ * D_;
  float* kp = Kp + (size_t)b * pStride + ((size_t)(pOff + i) * H_ + h) * D_;
  float* vp = Vp + (size_t)b * pStride + ((size_t)(pOff + i) * H_ + h) * D_;

  const float scale = 0.125f; // 1/sqrt(64)
  float sS = 0.f, s0 = 0.f, s1 = 0.f;
#pragma unroll
  for (int d = 0; d < D_; d += 4) {
    float4 a = *(const float4*)(k0 + d);
    float4 c = *(const float4*)(k1 + d);
    float px = 0.5f * (a.x + c.x), py = 0.5f * (a.y + c.y);
    float pz = 0.5f * (a.z + c.z), pw = 0.5f * (a.w + c.w);
    sS += px * px + py * py + pz * pz + pw * pw;
    s0 += px * a.x + py * a.y + pz * a.z + pw * a.w;
    s1 += px * c.x + py * c.y + pz * c.z + pw * c.w;
  }
  sS *= scale; s0 *= scale; s1 *= scale;
  float mx = fmaxf(sS, fmaxf(s0, s1));
  float eS = __expf(sS - mx), e0 = __expf(s0 - mx), e1 = __expf(s1 - mx);
  float dn = eS + e0 + e1 + 1e-9f;
  float wS = eS / dn, w0 = e0 / dn, w1 = e1 / dn;
  float c0 = 0.5f * wS + w0, c1 = 0.5f * wS + w1;
#pragma unroll
  for (int d = 0; d < D_; d += 4) {
    float4 a = *(const float4*)(k0 + d);
    float4 c = *(const float4*)(k1 + d);
    float4 r;
    r.x = c0 * a.x + c1 * c.x; r.y = c0 * a.y + c1 * c.y;
    r.z = c0 * a.z + c1 * c.z; r.w = c0 * a.w + c1 * c.w;
    *(float4*)(kp + d) = r;
    float4 av = *(const float4*)(v0 + d);
    float4 cv = *(const float4*)(v1 + d);
    float4 rv;
    rv.x = c0 * av.x + c1 * cv.x; rv.y = c0 * av.y + c1 * cv.y;
    rv.z = c0 * av.z + c1 * cv.z; rv.w = c0 * av.w + c1 * cv.w;
    *(float4*)(vp + d) = rv;
  }
}

// ---------------------------------------------------------------------------
// Phase 2: attention. One wave32 per (b, h, 16-query tile).
// Distinct-node columns j = 0..36 (padded to 48) for the tile:
//   0..15 : tile leaves, 16..23: level-1 nodes, 24..27: level-2,
//   28..29: level-3, 30..36: ancestor siblings at levels 4..10.
// ---------------------------------------------------------------------------
__device__ __forceinline__ int node_of(int j, int t) {
  if (j < 16) return t * 16 + j;                  // leaf
  if (j < 24) return 2048 + 8 * t + (j - 16);     // level 1
  if (j < 28) return 3072 + 4 * t + (j - 24);     // level 2
  if (j < 30) return 3584 + 2 * t + (j - 28);     // level 3
  if (j < 37) {                                    // levels 4..10 (shared)
    int lvl = j - 26;                              // 4..10
    int off = 2 * N_ - ((2 * N_) >> lvl);
    return off + ((t >> (lvl - 4)) ^ 1);
  }
  return 0; // pad columns (weights stay zero)
}

__global__ __launch_bounds__(32) void hsa_attn_kernel(
    const float* __restrict__ Q, const float* __restrict__ Kin,
    const float* __restrict__ Vin, const float* __restrict__ Kws,
    const float* __restrict__ Vws, float* __restrict__ out)
{
  const int t = blockIdx.x;      // query tile (16 queries)
  const int h = blockIdx.y;
  const int b = blockIdx.z;
  const int lane = threadIdx.x;  // 0..31
  const int m = lane & 15;
  const int half = lane >> 4;

  __shared__ unsigned long long kcol[48];  // global byte addresses of K rows
  __shared__ unsigned long long vcol[48];  // global byte addresses of V rows
  __shared__ float S[16][48];
  __shared__ float W[16][48];

  // per-column node row addresses (leaves from inputs, internal from ws)
  for (int j = lane; j < 48; j += 32) {
    int g = node_of(j, t);
    if (g < N_) {
      size_t off = (((size_t)b * N_ + g) * H_ + h) * D_;
      kcol[j] = (unsigned long long)(Kin + off);
      vcol[j] = (unsigned long long)(Vin + off);
    } else {
      size_t off = (((size_t)b * NINT_ + (g - N_)) * H_ + h) * D_;
      kcol[j] = (unsigned long long)(Kws + off);
      vcol[j] = (unsigned long long)(Vws + off);
    }
  }
  for (int x = lane; x < 16 * 48; x += 32) ((float*)W)[x] = 0.f;
  __syncthreads();

  // A fragments: Q tile 16x64. Lane holds row m, k = 4c + 2*half + {0,1}.
  const float* qrow = Q + (((size_t)b * N_ + t * 16 + m) * H_ + h) * D_;
  v2f qa[16];
#pragma unroll
  for (int c = 0; c < 16; ++c)
    qa[c] = *(const v2f*)(qrow + 4 * c + 2 * half);

  // S(16x48) = Q(16x64) x K^T(64x48) via v_wmma_f32_16x16x4_f32
  v8f acc[3];
#pragma unroll
  for (int ct = 0; ct < 3; ++ct) {
    v8f a = {};
    unsigned long long kp_ = kcol[ct * 16 + m] + (unsigned long long)(8 * half);
#pragma unroll
    for (int c = 0; c < 16; ++c) {
      v2f bf = load_g2(kp_ + (unsigned long long)(16 * c)); // B: col m, k=4c+2*half+{0,1}
      a = __builtin_amdgcn_wmma_f32_16x16x4_f32(false, qa[c], false, bf,
                                                (short)0, a, false, false);
    }
    acc[ct] = a;
  }
#pragma unroll
  for (int ct = 0; ct < 3; ++ct)
#pragma unroll
    for (int r = 0; r < 8; ++r)
      S[r + 8 * half][ct * 16 + m] = acc[ct][r];
  __syncthreads();

  // softmax over the 12 mapped columns per row; masked when ancestor is even
  if (lane < 16) {
    int row = lane;
    int jj[12]; bool mk[12];
    jj[0] = row;                   mk[0] = false;
    jj[1] = row ^ 1;               mk[1] = ((row & 1) == 0);
    jj[2] = 16 + ((row >> 1) ^ 1); mk[2] = (((row >> 1) & 1) == 0);
    jj[3] = 24 + ((row >> 2) ^ 1); mk[3] = (((row >> 2) & 1) == 0);
    jj[4] = 28 + ((row >> 3) ^ 1); mk[4] = (((row >> 3) & 1) == 0);
#pragma unroll
    for (int l = 5; l < 12; ++l) { jj[l] = 25 + l; mk[l] = (((t >> (l - 5)) & 1) == 0); }

    const float scale = 0.125f;
    float s[12];
    float mx = -INFINITY;
#pragma unroll
    for (int l = 0; l < 12; ++l) {
      s[l] = mk[l] ? -INFINITY : S[row][jj[l]] * scale;
      mx = fmaxf(mx, s[l]);
    }
    float e[12]; float sum = 0.f;
#pragma unroll
    for (int l = 0; l < 12; ++l) { e[l] = __expf(s[l] - mx); sum += e[l]; }
    float inv = 1.f / sum;
#pragma unroll
    for (int l = 0; l < 12; ++l) W[row][jj[l]] = e[l] * inv;
  }
  __syncthreads();

  // O(16x64) = W(16x48) x V(48x64)
  v8f o[4];
#pragma unroll
  for (int nt = 0; nt < 4; ++nt) {
    v8f a = {};
    int ncol = nt * 16 + m;
#pragma unroll
    for (int c = 0; c < 12; ++c) {
      int k0 = 4 * c + 2 * half;
      v2f af = *(const v2f*)(&W[m][k0]);              // A: row m, k = k0,k0+1
      v2f bf;
      bf.x = load_g1(vcol[k0]     + (unsigned long long)(4 * ncol));
      bf.y = load_g1(vcol[k0 + 1] + (unsigned long long)(4 * ncol));
      a = __builtin_amdgcn_wmma_f32_16x16x4_f32(false, af, false, bf,
                                                (short)0, a, false, false);
    }
    o[nt] = a;
  }

  float* op = out + (((size_t)b * N_ + t * 16 + 8 * half) * H_ + h) * D_;
#pragma unroll
  for (int nt = 0; nt < 4; ++nt)
#pragma unroll
    for (int r = 0; r < 8; ++r)
      op[(size_t)r * H_ * D_ + nt * 16 + m] = o[nt][r];
}

// ---------------------------------------------------------------------------
extern "C" void kernel_launch(void* const* d_in, const int* in_sizes, int n_in,
                              void* d_out, int out_size, void* d_ws, size_t ws_size,
                              hipStream_t stream) {
  const float* Q = (const float*)d_in[0];
  const float* K = (const float*)d_in[1];
  const float* V = (const float*)d_in[2];

  // workspace: internal tree nodes, K then V: 2 * B*(N-1)*H*D * 4B = 33.5 MB
  float* Kws = (float*)d_ws;
  float* Vws = Kws + (size_t)B_ * NINT_ * H_ * D_;

  const size_t inStride = (size_t)N_ * H_ * D_;     // input batch stride
  const size_t wsStride = (size_t)NINT_ * H_ * D_;  // ws batch stride

  // build tree, level by level (stream ordering provides dependencies)
  for (int lvl = 1; lvl <= 11; ++lvl) {
    int P = N_ >> lvl;
    int total = B_ * P * H_;
    int blk = 256, grd = (total + blk - 1) / blk;
    const float *kc, *vc; size_t cs; int cOff;
    if (lvl == 1) { kc = K; vc = V; cs = inStride; cOff = 0; }
    else {
      kc = Kws; vc = Vws; cs = wsStride;
      cOff = (2 * N_ - ((2 * N_) >> (lvl - 1))) - N_;
    }
    int pOff = (2 * N_ - ((2 * N_) >> lvl)) - N_;
    tree_level_kernel<<<grd, blk, 0, stream>>>(kc, vc, Kws, Vws, cs, wsStride,
                                               cOff, pOff, P);
  }

  // attention: one wave32 per (tile, head, batch)
  dim3 grid(N_ / 16, H_, B_);
  hsa_attn_kernel<<<grid, 32, 0, stream>>>(Q, K, V, Kws, Vws, (float*)d_out);
}